// NN_48696339202344
// MI455X (gfx1250) — compile-verified
//
#include <hip/hip_runtime.h>

// logits = x @ W ; x:[16384,8192] f32, W:[8192,16] f32, out:[16384,16] f32.
// Memory-bound (AI ~8 flop/B -> ~23us at 23.3 TB/s): stream x once with
// widest (b128) per-lane loads, W staged in LDS in a K-permuted fragment
// layout; math via V_WMMA_F32_16X16X4_F32. K-order inside each 8-wide chunk
// is permuted identically in A and B packing, which the contraction allows.

typedef __attribute__((ext_vector_type(2))) float v2f;
typedef __attribute__((ext_vector_type(4))) float v4f;
typedef __attribute__((ext_vector_type(8))) float v8f;

#define M_DIM 16384
#define K_DIM 8192
#define N_DIM 16

#define SEG_K 512                       // K-span of one LDS-resident W segment
#define NSEG (K_DIM / SEG_K)            // 16
#define WAVES_PER_BLOCK 4
#define THREADS (WAVES_PER_BLOCK * 32)  // 128 (wave32)
#define ROWS_PER_BLOCK (WAVES_PER_BLOCK * 16)  // 64
#define GRID (M_DIM / ROWS_PER_BLOCK)          // 256

__launch_bounds__(THREADS)
__global__ void neat_logits_wmma(const float* __restrict__ x,
                                 const float* __restrict__ W,
                                 float* __restrict__ out) {
    // Per 8-K superchunk c, lane l = hi*16+n holds the 16-byte B fragment
    //   { W[8c+4hi+0][n], W[8c+4hi+1][n], W[8c+4hi+2][n], W[8c+4hi+3][n] }
    // .xy feeds WMMA#1, .zw feeds WMMA#2 (matching the A b128 split below).
    __shared__ float ldsW[SEG_K * N_DIM];       // 32 KB

    const int tid  = threadIdx.x;
    const int wave = tid >> 5;
    const int lane = tid & 31;
    const int hi   = lane >> 4;     // half-wave: selects K group / M+8
    const int n    = lane & 15;     // column / row-in-tile

    const int row_base = blockIdx.x * ROWS_PER_BLOCK + wave * 16;
    // A-fragment source: lane supplies row (row_base + n),
    // contiguous K = 8c + 4*hi + {0,1,2,3} as one b128.
    const float* xrow = x + (size_t)(row_base + n) * K_DIM + 4 * hi;

    v8f acc = {0.f, 0.f, 0.f, 0.f, 0.f, 0.f, 0.f, 0.f};

    for (int seg = 0; seg < NSEG; ++seg) {
        const int segk = seg * SEG_K;

        // --- cooperative, coalesced load of W segment into permuted LDS ---
        for (int i = tid; i < SEG_K * N_DIM; i += THREADS) {
            const int k  = i >> 4;          // 0..SEG_K-1
            const int nn = i & 15;
            const float w = W[(size_t)(segk + k) * N_DIM + nn];
            const int c8 = k >> 3;          // 8-wide K superchunk
            const int r  = k & 7;
            const int l  = ((r >> 2) << 4) | nn;       // dest lane (hi,n)
            ldsW[c8 * 128 + l * 4 + (r & 3)] = w;
        }
        __syncthreads();

        // hint: pull the next K-segment of this lane's x row toward cache
        if (seg + 1 < NSEG)
            __builtin_prefetch(xrow + segk + SEG_K, 0, 1);

        const float* xa = xrow + segk;
        const float* ba = &ldsW[lane * 4];
#pragma unroll 8
        for (int c = 0; c < SEG_K / 8; ++c) {
            v4f a = *(const v4f*)(xa + c * 8);            // global b128 (x stream)
            v4f b = *(const v4f*)(ba + c * 128);          // ds b128 (W fragments)
            v2f a0 = {a.x, a.y}, a1 = {a.z, a.w};
            v2f b0 = {b.x, b.y}, b1 = {b.z, b.w};
            acc = __builtin_amdgcn_wmma_f32_16x16x4_f32(
                false, a0, false, b0, (short)0, acc, false, false);
            acc = __builtin_amdgcn_wmma_f32_16x16x4_f32(
                false, a1, false, b1, (short)0, acc, false, false);
        }
        __syncthreads();
    }

    // C/D layout: VGPR v, lanes 0-15 -> M=v, lanes 16-31 -> M=v+8; N = lane&15
    const int orow = row_base + hi * 8;
#pragma unroll
    for (int v = 0; v < 8; ++v)
        out[(size_t)(orow + v) * N_DIM + n] = acc[v];
}

extern "C" void kernel_launch(void* const* d_in, const int* in_sizes, int n_in,
                              void* d_out, int out_size, void* d_ws, size_t ws_size,
                              hipStream_t stream) {
    const float* x = (const float*)d_in[0];   // [16384, 8192] f32
    const float* W = (const float*)d_in[1];   // [8192, 16]   f32
    float* out = (float*)d_out;               // [16384, 16]  f32
    neat_logits_wmma<<<GRID, THREADS, 0, stream>>>(x, W, out);
}